// ChamferLoss_75548474736998
// MI455X (gfx1250) — compile-verified
//
#include <hip/hip_runtime.h>

typedef __attribute__((ext_vector_type(2))) float v2f;
typedef __attribute__((ext_vector_type(8))) float v8f;

#define NSEG 8  // ref-range segments per query tile -> 16K waves of parallelism

// Pack (x,y,z) -> (x,y,z, x^2+y^2+z^2) for both query and ref points.
__global__ void chamfer_pack(const float* __restrict__ qin,
                             const float* __restrict__ rin,
                             float4* __restrict__ q4,
                             float4* __restrict__ r4,
                             int nq, int nr) {
  int i = blockIdx.x * blockDim.x + threadIdx.x;
  if (i < nq) {
    float x = qin[3 * i], y = qin[3 * i + 1], z = qin[3 * i + 2];
    q4[i] = make_float4(x, y, z, __fmaf_rn(x, x, __fmaf_rn(y, y, z * z)));
  } else if (i < nq + nr) {
    int j = i - nq;
    float x = rin[3 * j], y = rin[3 * j + 1], z = rin[3 * j + 2];
    r4[j] = make_float4(x, y, z, __fmaf_rn(x, x, __fmaf_rn(y, y, z * z)));
  }
}

__device__ __forceinline__ void min8_asm(float* m, const v8f& d) {
#pragma unroll
  for (int rr = 0; rr < 8; ++rr) {
    float dv = d[rr];
    // raw v_min_num_f32: inputs are distances (never NaN), skip canonicalize
    asm("v_min_num_f32 %0, %0, %1" : "+v"(m[rr]) : "v"(dv));
  }
}

// One wave handles one 16-query tile x one ref segment.
// A (16x4 f32): row i = (-2qx, -2qy, -2qz, 1); lanes 0-15 hold K0/K1, lanes 16-31 K2/K3.
// B (4x16 f32): col j = (rx, ry, rz, r2); same half-wave K split.
// C init = q2 per row  =>  D = q2 - 2 q.r + r2 = squared-distance tile in one WMMA.
__global__ void chamfer_knn_wmma(const float4* __restrict__ q4,
                                 const float4* __restrict__ r4,
                                 float* __restrict__ rowmin,
                                 int tilesPerSeg) {
  const int lane = threadIdx.x & 31;
  // wave index is uniform within the wave: pin it (and everything derived,
  // incl. loop bounds) to SGPRs so the loop is scalar-controlled.
  const int w    = __builtin_amdgcn_readfirstlane(blockIdx.x * 8 + (threadIdx.x >> 5));
  const int tile = w / NSEG;
  const int seg  = w % NSEG;
  const int qbase = tile * 16;
  const int half = lane >> 4;   // 0: K=0,1 side   1: K=2,3 side
  const int l16  = lane & 15;

  // A operand (held in 2 VGPRs for the whole loop)
  float4 myq = q4[qbase + l16];
  v2f a;
  a.x = -2.0f * (half ? myq.z : myq.x);
  a.y = half ? 1.0f : (-2.0f * myq.y);

  // C init: register rr holds q2 of row (half*8 + rr), per C/D layout
  v8f cinit;
#pragma unroll
  for (int rr = 0; rr < 8; ++rr)
    cinit[rr] = q4[qbase + half * 8 + rr].w;

  float minv[8];
#pragma unroll
  for (int rr = 0; rr < 8; ++rr) minv[rr] = 3.0e38f;

  // packed ref: 2 float2 per point -> half 0 loads (x,y), half 1 loads (z,r2).
  // One 16-ref tile = 32 float2. Per-lane stream pointer for this segment.
  const float2* lanep = ((const float2*)r4)
                      + (size_t)(seg * tilesPerSeg) * 32 + (l16 * 2 + half);

  // Software-pipelined: next tile's B load is in flight during current WMMA.
  float2 bv = lanep[0];
  for (int t = 0; t < tilesPerSeg - 1; ++t) {
    float2 bn = lanep[(t + 1) * 32];
    v2f b; b.x = bv.x; b.y = bv.y;
    v8f d = __builtin_amdgcn_wmma_f32_16x16x4_f32(
        false, a, false, b, (short)0, cinit, false, false);
    min8_asm(minv, d);
    bv = bn;
  }
  {
    v2f b; b.x = bv.x; b.y = bv.y;
    v8f d = __builtin_amdgcn_wmma_f32_16x16x4_f32(
        false, a, false, b, (short)0, cinit, false, false);
    min8_asm(minv, d);
  }

  // min across the 16 columns held by the 16 lanes of each half-wave
#pragma unroll
  for (int m = 1; m <= 8; m <<= 1) {
#pragma unroll
    for (int rr = 0; rr < 8; ++rr)
      minv[rr] = fminf(minv[rr], __shfl_xor(minv[rr], m, 32));
  }

  if (l16 == 0) {
    float* outp = rowmin + ((size_t)tile * NSEG + seg) * 16 + half * 8;
#pragma unroll
    for (int rr = 0; rr < 8; ++rr) outp[rr] = minv[rr];
  }
}

// Per query: min over its NSEG segment results, clamp, block-sum (deterministic).
__global__ void chamfer_combine(const float* __restrict__ rowmin,
                                float* __restrict__ partial, int nq) {
  __shared__ float sm[256];
  int i = blockIdx.x * 256 + threadIdx.x;
  float v = 0.0f;
  if (i < nq) {
    int tile = i >> 4, row = i & 15;
    const float* p = rowmin + (size_t)tile * (NSEG * 16) + row;
    float m = p[0];
#pragma unroll
    for (int s = 1; s < NSEG; ++s) m = fminf(m, p[s * 16]);
    v = fmaxf(m, 0.0f);
  }
  sm[threadIdx.x] = v;
  __syncthreads();
  for (int k = 128; k > 0; k >>= 1) {
    if (threadIdx.x < k) sm[threadIdx.x] += sm[threadIdx.x + k];
    __syncthreads();
  }
  if (threadIdx.x == 0) partial[blockIdx.x] = sm[0];
}

__global__ void chamfer_final(const float* __restrict__ partial,
                              float* __restrict__ out, int nblocks, float inv) {
  __shared__ float sm[256];
  float s = 0.0f;
  for (int i = threadIdx.x; i < nblocks; i += 256) s += partial[i];
  sm[threadIdx.x] = s;
  __syncthreads();
  for (int k = 128; k > 0; k >>= 1) {
    if (threadIdx.x < k) sm[threadIdx.x] += sm[threadIdx.x + k];
    __syncthreads();
  }
  if (threadIdx.x == 0) out[0] = sm[0] * inv;
}

extern "C" void kernel_launch(void* const* d_in, const int* in_sizes, int n_in,
                              void* d_out, int out_size, void* d_ws, size_t ws_size,
                              hipStream_t stream) {
  const float* q = (const float*)d_in[0];
  const float* r = (const float*)d_in[1];
  const int nq = in_sizes[0] / 3;   // 32768 query points (B*N)
  const int nr = in_sizes[1] / 3;   // 8192 ref points
  const int nTilesQ = nq / 16;      // 2048
  const int nTilesR = nr / 16;      // 512
  const int tilesPerSeg = nTilesR / NSEG;

  // workspace layout
  char* ws = (char*)d_ws;
  float4* q4 = (float4*)ws;                       ws += (size_t)nq * 16;
  float4* r4 = (float4*)ws;                       ws += (size_t)nr * 16;
  float* rowmin = (float*)ws;                     ws += (size_t)nTilesQ * NSEG * 16 * sizeof(float);
  float* partial = (float*)ws;

  const int total = nq + nr;
  chamfer_pack<<<(total + 255) / 256, 256, 0, stream>>>(q, r, q4, r4, nq, nr);

  const int nWaves = nTilesQ * NSEG;              // 16384 waves
  chamfer_knn_wmma<<<nWaves / 8, 256, 0, stream>>>(q4, r4, rowmin, tilesPerSeg);

  const int nCombBlocks = (nq + 255) / 256;       // 128
  chamfer_combine<<<nCombBlocks, 256, 0, stream>>>(rowmin, partial, nq);
  chamfer_final<<<1, 256, 0, stream>>>(partial, (float*)d_out, nCombBlocks,
                                       1.0f / (float)nq);
}